// MultiUniverseToposAttention_70145405878335
// MI455X (gfx1250) — compile-verified
//
#include <hip/hip_runtime.h>
#include <math.h>

// MI455X / gfx1250, wave32. WMMA f32 16x16x4 for all GEMM stages.
// Weights pre-sigmoided once; truth stage tiled 16q/block for 16x K-reuse;
// gate-combine fused into the attn@V epilogue.

typedef __attribute__((ext_vector_type(2))) float v2f;
typedef __attribute__((ext_vector_type(8))) float v8f;

#define S_LEN 512
#define DMODEL 512
#define NHEAD 8
#define DHEAD 64
#define HALF 32
#define QT 16  // q-tile for truth kernel

__device__ __forceinline__ float sigm(float x) { return 1.0f / (1.0f + expf(-x)); }

// ---------------------------------------------------------------------------
// Pre-activate weights: dst[z] = sigmoid(W[z]) for z in {Wq,Wk,Wv,Wo}
// ---------------------------------------------------------------------------
__global__ void presig_kernel(const float* __restrict__ W0,
                              const float* __restrict__ W1,
                              const float* __restrict__ W2,
                              const float* __restrict__ W3,
                              float* __restrict__ dst) {
  int z = blockIdx.y;
  const float* src = (z == 0) ? W0 : (z == 1) ? W1 : (z == 2) ? W2 : W3;
  int i = blockIdx.x * blockDim.x + threadIdx.x;
  dst[(size_t)z * DMODEL * DMODEL + i] = sigm(src[i]);
}

// ---------------------------------------------------------------------------
// Router: logits = x @ sig(Wr)^T, softmax over 8 heads, top-2, normalized gate
// ---------------------------------------------------------------------------
__global__ void router_kernel(const float* __restrict__ x,
                              const float* __restrict__ Wr,
                              float* __restrict__ gate) {
  int s = blockIdx.x;
  int lane = threadIdx.x & 31;
  int wave = threadIdx.x >> 5;  // head index 0..7
  __shared__ float logits[NHEAD];
  const float* xr = x + (size_t)s * DMODEL;
  const float* wr = Wr + (size_t)wave * DMODEL;
  float acc = 0.0f;
  for (int d = lane; d < DMODEL; d += 32) acc += xr[d] * sigm(wr[d]);
  for (int off = 16; off > 0; off >>= 1) acc += __shfl_xor(acc, off, 32);
  if (lane == 0) logits[wave] = acc;
  __syncthreads();
  if (threadIdx.x == 0) {
    float m = logits[0];
    for (int h = 1; h < NHEAD; ++h) m = fmaxf(m, logits[h]);
    float e[NHEAD];
    float Z = 0.0f;
    for (int h = 0; h < NHEAD; ++h) { e[h] = expf(logits[h] - m); Z += e[h]; }
    int i0 = 0;
    for (int h = 1; h < NHEAD; ++h) if (e[h] > e[i0]) i0 = h;
    int i1 = -1;
    for (int h = 0; h < NHEAD; ++h) {
      if (h == i0) continue;
      if (i1 < 0 || e[h] > e[i1]) i1 = h;
    }
    float p0 = e[i0] / Z, p1 = e[i1] / Z;
    float inv = 1.0f / (p0 + p1 + 1e-9f);
    for (int h = 0; h < NHEAD; ++h) gate[s * NHEAD + h] = 0.0f;
    gate[s * NHEAD + i0] = p0 * inv;
    gate[s * NHEAD + i1] = p1 * inv;
  }
}

// ---------------------------------------------------------------------------
// C = A(512xK) @ Bw^T, Bw pre-activated NxK row-major; C row-major 512xN.
// One wave computes a 64(M) x 16(N) strip: 4 accumulators share one B frag.
// blockIdx.z selects weight slab (Bw + z*DM*DM) and output slab (C + z*S*DM).
// ---------------------------------------------------------------------------
__global__ void gemm_bw_wmma(const float* __restrict__ A,
                             const float* __restrict__ Bw,
                             float* __restrict__ C) {
  const int K = DMODEL, N = DMODEL;
  const float* B = Bw + (size_t)blockIdx.z * DMODEL * DMODEL;
  float* Cz = C + (size_t)blockIdx.z * S_LEN * DMODEL;
  int lane = threadIdx.x & 31;
  int l15 = lane & 15;
  int half = lane >> 4;  // 0 or 1
  int n0 = blockIdx.x * 16;
  int m0 = blockIdx.y * 64;
  const float* brow = B + (size_t)(n0 + l15) * K;  // B[k][n] = Bw[n][k]
  const float* ar[4];
#pragma unroll
  for (int t = 0; t < 4; ++t) ar[t] = A + (size_t)(m0 + t * 16 + l15) * K;
  v8f acc[4] = {{}, {}, {}, {}};
  for (int k0 = 0; k0 < K; k0 += 4) {
    int kk = k0 + 2 * half;
    v2f b;
    b.x = brow[kk];
    b.y = brow[kk + 1];
#pragma unroll
    for (int t = 0; t < 4; ++t) {
      v2f a;
      a.x = ar[t][kk];
      a.y = ar[t][kk + 1];
      acc[t] = __builtin_amdgcn_wmma_f32_16x16x4_f32(false, a, false, b,
                                                     (short)0, acc[t], false,
                                                     false);
    }
  }
  float* crow = Cz + n0 + l15;
#pragma unroll
  for (int t = 0; t < 4; ++t)
#pragma unroll
    for (int r = 0; r < 8; ++r)
      crow[(size_t)(m0 + t * 16 + r + 8 * half) * N] = acc[t][r];
}

// ---------------------------------------------------------------------------
// RoPE (interleaved pairs) + sigmoid, in place on Q (blockIdx.y==0) or K (==1)
// ---------------------------------------------------------------------------
__global__ void rope_sig_kernel(float* __restrict__ Q, float* __restrict__ K,
                                const float* __restrict__ fc,
                                const float* __restrict__ fs) {
  int s = blockIdx.x;
  int p = threadIdx.x;         // pair index 0..255 across DM
  int i = p & (HALF - 1);      // pair index within head (0..31)
  float c = fc[s * HALF + i];
  float sn = fs[s * HALF + i];
  float* buf = (blockIdx.y == 0) ? Q : K;
  float* v = buf + (size_t)s * DMODEL + 2 * p;
  float v0 = v[0], v1 = v[1];
  v[0] = sigm(v0 * c - v1 * sn);
  v[1] = sigm(v0 * sn + v1 * c);
}

// ---------------------------------------------------------------------------
// truth[h,q,k] = mean_d clip(1 - Qs[q,h,d] + Ks[k,h,d], 0, 1); denom = row sum
// Tiled: one block = (h, 16 q's). Q tile staged in LDS; each thread streams
// 2 k-rows, keeping each K chunk in registers and reusing it across 16 q's
// (16x reduction in K traffic). Row sums via ds_add_f32 shared atomics.
// ---------------------------------------------------------------------------
__global__ void truth_kernel(const float* __restrict__ Qs,
                             const float* __restrict__ Ks,
                             float* __restrict__ truth,
                             float* __restrict__ denom) {
  int h = blockIdx.y;
  int q0 = blockIdx.x * QT;
  int tid = threadIdx.x;
  __shared__ float qs[QT][DHEAD];  // 4 KB
  __shared__ float dsum[QT];
  for (int i = tid; i < QT * DHEAD; i += 256) {
    int q = i >> 6, d = i & 63;
    qs[q][d] = Qs[(size_t)(q0 + q) * DMODEL + h * DHEAD + d];
  }
  if (tid < QT) dsum[tid] = 0.0f;
  __syncthreads();

  float acc[QT];
#pragma unroll
  for (int q = 0; q < QT; ++q) acc[q] = 0.0f;

  for (int kb = 0; kb < S_LEN; kb += 256) {
    int k = kb + tid;
    const float* krow = Ks + (size_t)k * DMODEL + h * DHEAD;
    float tq[QT];
#pragma unroll
    for (int q = 0; q < QT; ++q) tq[q] = 0.0f;
#pragma unroll
    for (int dc = 0; dc < DHEAD; dc += 16) {
      float kr[16];
#pragma unroll
      for (int d = 0; d < 16; ++d) kr[d] = krow[dc + d];
#pragma unroll
      for (int q = 0; q < QT; ++q) {
        float t = 0.0f;
#pragma unroll
        for (int d = 0; d < 16; ++d) {
          float v = 1.0f - qs[q][dc + d] + kr[d];
          t += fminf(fmaxf(v, 0.0f), 1.0f);
        }
        tq[q] += t;
      }
    }
#pragma unroll
    for (int q = 0; q < QT; ++q) {
      float t = tq[q] * (1.0f / 64.0f);
      truth[((size_t)h * S_LEN + q0 + q) * S_LEN + k] = t;  // coalesced in k
      acc[q] += t;
    }
  }
#pragma unroll
  for (int q = 0; q < QT; ++q) atomicAdd(&dsum[q], acc[q]);
  __syncthreads();
  if (tid < QT) denom[h * S_LEN + q0 + tid] = dsum[tid];
}

// ---------------------------------------------------------------------------
// fin[m, h*64+n] = gate[m,h]/(denom[h,m]+1e-9) * sum_k truth[h,m,k]*V[k,h*64+n]
// 64(M) x 16(N) strip per wave; B frag shared across 4 accumulators.
// Gate-combine fused into the store epilogue.
// ---------------------------------------------------------------------------
__global__ void av_gemm_wmma(const float* __restrict__ truth,
                             const float* __restrict__ V,
                             const float* __restrict__ denom,
                             const float* __restrict__ gate,
                             float* __restrict__ fin) {
  int h = blockIdx.z;
  int lane = threadIdx.x & 31;
  int l15 = lane & 15;
  int half = lane >> 4;
  int n0 = blockIdx.x * 16;
  int m0 = blockIdx.y * 64;
  const float* ar[4];
#pragma unroll
  for (int t = 0; t < 4; ++t)
    ar[t] = truth + ((size_t)h * S_LEN + m0 + t * 16 + l15) * S_LEN;
  const float* vcol = V + h * DHEAD + n0 + l15;  // B[k][n] = V[k*DM + h*64 + n]
  v8f acc[4] = {{}, {}, {}, {}};
  for (int k0 = 0; k0 < S_LEN; k0 += 4) {
    int kk = k0 + 2 * half;
    v2f b;
    b.x = vcol[(size_t)kk * DMODEL];
    b.y = vcol[(size_t)(kk + 1) * DMODEL];
#pragma unroll
    for (int t = 0; t < 4; ++t) {
      v2f a;
      a.x = ar[t][kk];
      a.y = ar[t][kk + 1];
      acc[t] = __builtin_amdgcn_wmma_f32_16x16x4_f32(false, a, false, b,
                                                     (short)0, acc[t], false,
                                                     false);
    }
  }
#pragma unroll
  for (int t = 0; t < 4; ++t)
#pragma unroll
    for (int r = 0; r < 8; ++r) {
      int m = m0 + t * 16 + r + 8 * half;
      float sc = gate[m * NHEAD + h] / (denom[h * S_LEN + m] + 1e-9f);
      fin[(size_t)m * DMODEL + h * DHEAD + n0 + l15] = acc[t][r] * sc;
    }
}

// ---------------------------------------------------------------------------
extern "C" void kernel_launch(void* const* d_in, const int* in_sizes, int n_in,
                              void* d_out, int out_size, void* d_ws,
                              size_t ws_size, hipStream_t stream) {
  (void)in_sizes; (void)n_in; (void)out_size; (void)ws_size;
  const float* x  = (const float*)d_in[0];
  const float* fc = (const float*)d_in[1];
  const float* fs = (const float*)d_in[2];
  const float* Wq = (const float*)d_in[3];
  const float* Wk = (const float*)d_in[4];
  const float* Wv = (const float*)d_in[5];
  const float* Wo = (const float*)d_in[6];
  const float* Wr = (const float*)d_in[7];
  float* out = (float*)d_out;

  const size_t WSZ = (size_t)DMODEL * DMODEL;   // 262144
  const size_t XSZ = (size_t)S_LEN * DMODEL;    // 262144

  float* ws = (float*)d_ws;
  float* gate  = ws;                            // 512*8
  float* sW    = gate + S_LEN * NHEAD;          // 4 * 512*512 (sWq,sWk,sWv,sWo)
  float* QKV   = sW + 4 * WSZ;                  // 3 * 512*512 (Q,K,V contiguous)
  float* truth = QKV + 3 * XSZ;                 // 8*512*512
  float* denom = truth + (size_t)NHEAD * S_LEN * S_LEN;  // 8*512
  float* fin   = denom + NHEAD * S_LEN;         // 512*512

  float* Q = QKV;
  float* K = QKV + XSZ;
  float* V = QKV + 2 * XSZ;
  float* sWo = sW + 3 * WSZ;

  // 1) sigmoid of the four big weight matrices
  presig_kernel<<<dim3((DMODEL * DMODEL) / 256, 4), 256, 0, stream>>>(
      Wq, Wk, Wv, Wo, sW);

  // 2) router gate
  router_kernel<<<S_LEN, 256, 0, stream>>>(x, Wr, gate);

  // 3) Q/K/V projections (z = 0/1/2 selects weight + output slab)
  gemm_bw_wmma<<<dim3(DMODEL / 16, S_LEN / 64, 3), 32, 0, stream>>>(x, sW, QKV);

  // 4) RoPE + sigmoid on Q and K
  rope_sig_kernel<<<dim3(S_LEN, 2), 256, 0, stream>>>(Q, K, fc, fs);

  // 5) clipped-similarity attention scores + row sums (q-tiled, K-reuse x16)
  truth_kernel<<<dim3(S_LEN / QT, NHEAD), 256, 0, stream>>>(Q, K, truth, denom);

  // 6) attn @ V with row normalization + top-2 gate fused into epilogue
  av_gemm_wmma<<<dim3(DHEAD / 16, S_LEN / 64, NHEAD), 32, 0, stream>>>(
      truth, V, denom, gate, fin);

  // 7) output projection
  gemm_bw_wmma<<<dim3(DMODEL / 16, S_LEN / 64, 1), 32, 0, stream>>>(fin, sWo,
                                                                    out);
}